// MSE_loss_edges_47691316855372
// MI455X (gfx1250) — compile-verified
//
#include <hip/hip_runtime.h>

// Problem constants (reference: B,C,H,W = 16,1,1024,1024)
#define HH 1024
#define WW 1024
#define BB 16
#define ROWS 16                 // output rows per block
#define TROWS (ROWS + 8)        // + 4-row halo each side (9x9 max window)
#define NTILES (HH / ROWS)      // 64 row-tiles per image
#define BLOCK 256               // 8 waves (wave32)
#define VROW (WW + 8)           // vmin row with 4-col halo each side

#if __has_builtin(__builtin_amdgcn_global_load_async_to_lds_b128)
#define USE_ASYNC 1
#else
#define USE_ASYNC 0
#endif

typedef float v2f __attribute__((ext_vector_type(2)));
typedef float v8f __attribute__((ext_vector_type(8)));
typedef int   v4i __attribute__((vector_size(4 * sizeof(int))));

typedef __attribute__((address_space(1))) v4i* gv4i_ptr;  // global int4*
typedef __attribute__((address_space(3))) v4i* lv4i_ptr;  // LDS int4*

__device__ __forceinline__ float f3min(float a, float b, float c) {
  return fminf(fminf(a, b), c);
}
__device__ __forceinline__ float4 f4min(float4 a, float4 b) {
  float4 o;
  o.x = fminf(a.x, b.x); o.y = fminf(a.y, b.y);
  o.z = fminf(a.z, b.z); o.w = fminf(a.w, b.w);
  return o;
}

// ---------------------------------------------------------------------------
// Main kernel: fused 4x erosion (separable min) + weighted MSE partial sums.
// One block = one batch image x 16-row tile. y_true tile staged in LDS via
// async global->LDS loads (CDNA5 ASYNCcnt path).
// ---------------------------------------------------------------------------
extern "C" __global__ void __launch_bounds__(BLOCK)
erode_mse_kernel(const float* __restrict__ y_pred,
                 const float* __restrict__ y_true,
                 float* __restrict__ partials) {
  extern __shared__ float smem[];
  float* s_true = smem;                    // TROWS * WW
  float* s_vmin = smem + TROWS * WW;       // 4 * VROW
  float* s_red  = s_vmin + 4 * VROW;       // BLOCK/32

  const int tid   = threadIdx.x;
  const int tile  = blockIdx.x;            // 0..63
  const int batch = blockIdx.y;            // 0..15
  const int r0    = tile * ROWS;
  const int xc    = tid * 4;               // this thread's 4 columns

  const size_t img_base = (size_t)batch * HH * WW;

  // ---- stage y_true rows r0-4 .. r0+ROWS+3 into LDS (zero-pad OOB rows) ----
  for (int tr = 0; tr < TROWS; ++tr) {
    const int g = r0 - 4 + tr;
    float* dst = &s_true[tr * WW + xc];
    if (g >= 0 && g < HH) {
      const float* src = y_true + img_base + (size_t)g * WW + xc;
#if USE_ASYNC
      __builtin_amdgcn_global_load_async_to_lds_b128(
          (gv4i_ptr)(void*)(const_cast<float*>(src)),
          (lv4i_ptr)(void*)dst,
          0, 0);
#else
      *(float4*)dst = *(const float4*)src;
#endif
    } else {
      float4 z = {0.f, 0.f, 0.f, 0.f};
      *(float4*)dst = z;
    }
  }

  // zero the vmin column halos once (cols -4..-1 and WW..WW+3, 4 radii)
  if (tid < 32) {
    const int k = tid >> 3;                              // radius array 0..3
    const int h = tid & 7;                               // 0..7
    const int col = (h < 4) ? h : (WW + 4 + (h - 4));    // 0..3, 1028..1031
    s_vmin[k * VROW + col] = 0.0f;
  }

#if USE_ASYNC
#if __has_builtin(__builtin_amdgcn_s_wait_asynccnt)
  __builtin_amdgcn_s_wait_asynccnt(0);
#else
  asm volatile("s_wait_asynccnt 0" ::: "memory");
#endif
#endif
  __syncthreads();

  float lsum = 0.0f;

  for (int i = 0; i < ROWS; ++i) {
    const int ti = i + 4;          // tile row of the output row
    const int y  = r0 + i;         // global row

    // ---- vertical mins over rows ti-4..ti+4 for this thread's 4 columns ----
    float4 r[9];
#pragma unroll
    for (int d = 0; d < 9; ++d)
      r[d] = *(const float4*)&s_true[(ti - 4 + d) * WW + xc];

    const float4 v1 = f4min(f4min(r[3], r[4]), r[5]);
    const float4 v2 = f4min(v1, f4min(r[2], r[6]));
    const float4 v3 = f4min(v2, f4min(r[1], r[7]));
    const float4 v4 = f4min(v3, f4min(r[0], r[8]));

    *(float4*)&s_vmin[0 * VROW + 4 + xc] = v1;
    *(float4*)&s_vmin[1 * VROW + 4 + xc] = v2;
    *(float4*)&s_vmin[2 * VROW + 4 + xc] = v3;
    *(float4*)&s_vmin[3 * VROW + 4 + xc] = v4;

    // prefetch y_pred for this row while waiting on the barrier
    const float4 p4 = *(const float4*)(y_pred + img_base + (size_t)y * WW + xc);
    const float tv[4] = {r[4].x, r[4].y, r[4].z, r[4].w};
    const float pv[4] = {p4.x, p4.y, p4.z, p4.w};

    __syncthreads();

    // ---- horizontal windows: 12 contiguous vmin values per radius ----
    float w1[12], w2[12], w3[12], w4[12];
#pragma unroll
    for (int t = 0; t < 3; ++t) {
      float4 qa = *(const float4*)&s_vmin[0 * VROW + xc + t * 4];
      float4 qb = *(const float4*)&s_vmin[1 * VROW + xc + t * 4];
      float4 qc = *(const float4*)&s_vmin[2 * VROW + xc + t * 4];
      float4 qd = *(const float4*)&s_vmin[3 * VROW + xc + t * 4];
      w1[t*4+0]=qa.x; w1[t*4+1]=qa.y; w1[t*4+2]=qa.z; w1[t*4+3]=qa.w;
      w2[t*4+0]=qb.x; w2[t*4+1]=qb.y; w2[t*4+2]=qb.z; w2[t*4+3]=qb.w;
      w3[t*4+0]=qc.x; w3[t*4+1]=qc.y; w3[t*4+2]=qc.z; w3[t*4+3]=qc.w;
      w4[t*4+0]=qd.x; w4[t*4+1]=qd.y; w4[t*4+2]=qd.z; w4[t*4+3]=qd.w;
    }

#pragma unroll
    for (int j = 0; j < 4; ++j) {
      const float e1 = f3min(w1[j+3], w1[j+4], w1[j+5]);
      const float e2 = fminf(f3min(w2[j+2], w2[j+3], w2[j+4]),
                             fminf(w2[j+5], w2[j+6]));
      const float e3 = fminf(f3min(w3[j+1], w3[j+2], w3[j+3]),
                             fminf(f3min(w3[j+4], w3[j+5], w3[j+6]), w3[j+7]));
      const float e4 = fminf(fminf(f3min(w4[j+0], w4[j+1], w4[j+2]),
                                   f3min(w4[j+3], w4[j+4], w4[j+5])),
                             f3min(w4[j+6], w4[j+7], w4[j+8]));
      const float t  = tv[j];
      const float circ = 0.4f * (t  - e1) + 0.5f * (e1 - e2)
                       + 0.6f * (e2 - e3) + 0.7f * (e3 - e4);
      const float wgt = (circ == 0.0f) ? 0.8f : circ;
      const float df  = pv[j] - t;
      lsum += df * df * wgt;
    }

    __syncthreads();   // protect s_vmin before next row overwrites it
  }

  // ---- deterministic block reduction ----
#pragma unroll
  for (int off = 16; off > 0; off >>= 1)
    lsum += __shfl_down(lsum, off, 32);
  if ((tid & 31) == 0) s_red[tid >> 5] = lsum;
  __syncthreads();
  if (tid == 0) {
    float s = 0.0f;
#pragma unroll
    for (int wv = 0; wv < BLOCK / 32; ++wv) s += s_red[wv];
    partials[batch * NTILES + tile] = s;
  }
}

// ---------------------------------------------------------------------------
// Finalize: reduce 16x64 partials on the matrix core.
// A (16x4 f32) = per-batch partial chunk, B = ones => D rows = per-batch sums,
// accumulated through C across 16 chunks. Single wave, EXEC all-ones.
// ---------------------------------------------------------------------------
extern "C" __global__ void __launch_bounds__(32)
finalize_kernel(const float* __restrict__ partials, float* __restrict__ out) {
  const int lane = threadIdx.x;          // 0..31
  const int m    = lane & 15;            // batch row
  const int koff = (lane >> 4) << 1;     // K 0/1 (lanes 0-15) or 2/3 (16-31)

  v8f acc = {};
  v2f ones; ones[0] = 1.0f; ones[1] = 1.0f;

#pragma unroll
  for (int c = 0; c < 16; ++c) {
    v2f a;
    const int base = m * NTILES + c * 4 + koff;
    a[0] = partials[base];
    a[1] = partials[base + 1];
#if __has_builtin(__builtin_amdgcn_wmma_f32_16x16x4_f32)
    acc = __builtin_amdgcn_wmma_f32_16x16x4_f32(
        /*neg_a=*/false, a, /*neg_b=*/false, ones,
        /*c_mod=*/(short)0, acc, /*reuse_a=*/false, /*reuse_b=*/false);
#else
    acc[m & 7] += a[0] + a[1];   // compile-safe fallback (layout differs)
#endif
  }

  // acc[v]: lanes 0-15 -> batchsum[v], lanes 16-31 -> batchsum[8+v]
  float half8 = 0.0f;
#pragma unroll
  for (int v = 0; v < 8; ++v) half8 += acc[v];
  const float other = __shfl_down(half8, 16, 32);

  const float inv_hw  = 1.0f / (float)(HH * WW);
  const float inv_all = 1.0f / ((float)BB * (float)(HH * WW));

  if (lane == 0) {
#pragma unroll
    for (int v = 0; v < 8; ++v) out[v] = acc[v] * inv_hw;
    out[16] = (half8 + other) * inv_all;     // global mean
  }
  if (lane == 16) {
#pragma unroll
    for (int v = 0; v < 8; ++v) out[8 + v] = acc[v] * inv_hw;
  }
}

// ---------------------------------------------------------------------------
extern "C" void kernel_launch(void* const* d_in, const int* in_sizes, int n_in,
                              void* d_out, int out_size, void* d_ws, size_t ws_size,
                              hipStream_t stream) {
  const float* y_pred = (const float*)d_in[0];
  const float* y_true = (const float*)d_in[1];
  float* out      = (float*)d_out;          // 17 floats: 16 batch means + mean
  float* partials = (float*)d_ws;           // 16*64 partial sums (4 KB)

  const dim3 grid(NTILES, BB);
  const size_t shmem =
      (size_t)(TROWS * WW + 4 * VROW + BLOCK / 32) * sizeof(float); // ~112 KB

  erode_mse_kernel<<<grid, BLOCK, shmem, stream>>>(y_pred, y_true, partials);
  finalize_kernel<<<1, 32, 0, stream>>>(partials, out);
}